// FIRFilter_738734375484
// MI455X (gfx1250) — compile-verified
//
#include <hip/hip_runtime.h>

// FIR depthwise causal filter on MI455X (gfx1250).
// out[b,t,c] = sum_{i<16} w[i,c] * x[b, t-15+i, c]   (zero-padded past)
//
// Memory-bound (33.6 MB @ 23.3 TB/s ~ 1.44 us). Strategy:
//   - one workgroup (1024 thr = 32 waves) per (batch, 256-timestep chunk)
//   - coalesced global->LDS staging of x (read exactly once)
//   - per wave (= one channel): Toeplitz weights in regs as WMMA B operand,
//     x window from LDS as A operand, chain of 8 v_wmma_f32_16x16x4_f32
//     accumulating the 16x16 output tile (t = t0 + m*16 + n)
//   - D regs -> LDS (buffer reuse) -> coalesced global stores (written once)

typedef float v2f __attribute__((ext_vector_type(2)));
typedef float v8f __attribute__((ext_vector_type(8)));

#define NB 64
#define NT 2048
#define NC 32
#define NF 16
#define TCH 256              // timesteps per workgroup chunk (one 16x16 tile per wave)
#define XS_ROW 289           // LDS row stride for x stage (289%64==33, odd -> conflict-free staging writes)
#define OS_ROW 257           // LDS row stride for out stage (257%64==1 -> conflict-free coalesced reads)

__global__ __launch_bounds__(1024)
void fir_wmma_kernel(const float* __restrict__ x,
                     const float* __restrict__ fw,
                     float* __restrict__ out) {
    // Shared buffer: x-stage (32*289 floats = 36992B) then reused as out-stage (32*257).
    __shared__ float lds[NC * XS_ROW];

    const int tid  = threadIdx.x;
    const int t0   = blockIdx.x * TCH;   // chunk start
    const int b    = blockIdx.y;         // batch

    // ---------------- Stage x[b, t0-15 .. t0+256] for all 32 channels ----------------
    // 272 timesteps * 32 channels = 8704 floats, loaded in memory order (coalesced).
    // LDS index: c*289 + (tt&15)*17 + (tt>>4)  -- swizzle so that
    //   (a) staging writes (c = lane) are conflict-free (stride 289 ~ 33 mod 64)
    //   (b) A-operand reads (tt = m*16 + const, m = lane) are consecutive addresses.
    const long xbase = (long)b * NT * NC;
    #pragma unroll
    for (int it = 0; it < 9; ++it) {
        int f = tid + it * 1024;
        if (f < 272 * NC) {
            int cc = f & (NC - 1);
            int tt = f >> 5;
            int t  = t0 - (NF - 1) + tt;
            float v = 0.0f;
            if (t >= 0 && t < NT) v = x[xbase + (long)t * NC + cc];
            lds[cc * XS_ROW + (tt & 15) * 17 + (tt >> 4)] = v;
        }
    }
    __syncthreads();

    // ---------------- Per-wave WMMA: wave w handles channel c = w ----------------
    const int lane = tid & 31;
    const int c    = tid >> 5;     // wave id = channel
    const int mn   = lane & 15;    // lane's M row (for A/D) and N col (for B)
    const int hi   = lane >> 4;    // upper half-wave -> K offset +2

    // B operand: banded Toeplitz of the 16 taps, B[s,n] = w[s-n, c] (0 outside band).
    // VGPR j of K-block kb holds k = 4*kb + 2*hi + j  (ISA 7.12.2 layout).
    v2f bw[8];
    #pragma unroll
    for (int kb = 0; kb < 8; ++kb) {
        int s0 = 4 * kb + 2 * hi;
        int i0 = s0 - mn;
        int i1 = i0 + 1;
        float b0 = (i0 >= 0 && i0 < NF) ? fw[i0 * NC + c] : 0.0f;
        float b1 = (i1 >= 0 && i1 < NF) ? fw[i1 * NC + c] : 0.0f;
        bw[kb] = v2f{b0, b1};
    }

    // A operand from LDS + chained accumulation: K = 32 (band limits it to 31).
    v8f acc = {0.f, 0.f, 0.f, 0.f, 0.f, 0.f, 0.f, 0.f};
    #pragma unroll
    for (int kb = 0; kb < 8; ++kb) {
        int tt0 = mn * 16 + 4 * kb + 2 * hi;          // A[m, s] with s = 4kb+2hi(+1)
        int tt1 = tt0 + 1;
        float a0 = lds[c * XS_ROW + (tt0 & 15) * 17 + (tt0 >> 4)];
        float a1 = lds[c * XS_ROW + (tt1 & 15) * 17 + (tt1 >> 4)];
        v2f a = v2f{a0, a1};
        // (neg_a, A, neg_b, B, c_mod, C, reuse_a, reuse_b)
        acc = __builtin_amdgcn_wmma_f32_16x16x4_f32(
            false, a, false, bw[kb], (short)0, acc, false, false);
    }
    __syncthreads();   // all A-reads done before the buffer is reused for output

    // ---------------- D -> LDS out-stage ----------------
    // D VGPR d holds M = d + 8*hi (lanes 0-15 / 16-31), N = mn; t_local = M*16 + N.
    #pragma unroll
    for (int d = 0; d < 8; ++d) {
        int tl = (d + 8 * hi) * 16 + mn;
        lds[c * OS_ROW + tl] = acc[d];
    }
    __syncthreads();

    // ---------------- Coalesced global store (memory order over [t,c]) ----------------
    const long obase = (long)b * NT * NC + (long)t0 * NC;
    #pragma unroll
    for (int it = 0; it < 8; ++it) {
        int f  = tid + it * 1024;          // 8192 floats = 256 t * 32 c
        int cc = f & (NC - 1);
        int tt = f >> 5;
        out[obase + (long)tt * NC + cc] = lds[cc * OS_ROW + tt];
    }
}

extern "C" void kernel_launch(void* const* d_in, const int* in_sizes, int n_in,
                              void* d_out, int out_size, void* d_ws, size_t ws_size,
                              hipStream_t stream) {
    const float* x  = (const float*)d_in[0];   // [64, 2048, 32] f32
    const float* fw = (const float*)d_in[1];   // [16, 32] f32
    float* out      = (float*)d_out;           // [64, 2048, 32] f32
    (void)in_sizes; (void)n_in; (void)out_size; (void)d_ws; (void)ws_size;

    dim3 grid(NT / TCH, NB);   // (8, 64)
    dim3 block(1024);
    fir_wmma_kernel<<<grid, block, 0, stream>>>(x, fw, out);
}